// WindowAttention_79190607004165
// MI455X (gfx1250) — compile-verified
//
#include <hip/hip_runtime.h>

typedef float v2f __attribute__((ext_vector_type(2)));
typedef float v8f __attribute__((ext_vector_type(8)));

#define B_   2
#define H_   16
#define N_   2048
#define D_   64
#define GNUM 8
#define WIN  64
#define NT   (N_/16)   // 128 key tiles
#define QB   (N_/16)   // 128 query blocks per (b,h)

// D = A(16x4) * B(4x16) + C, f32.  A/B: 2 VGPR/lane, C/D: 8 VGPR/lane.
static __device__ __forceinline__ v8f wmma_f32(v2f a, v2f b, v8f c) {
  return __builtin_amdgcn_wmma_f32_16x16x4_f32(
      /*neg_a=*/false, a, /*neg_b=*/false, b,
      /*c_mod=*/(short)0, c, /*reuse_a=*/false, /*reuse_b=*/false);
}

__global__ __launch_bounds__(32) void wattn_kernel(
    const float* __restrict__ Q, const float* __restrict__ K,
    const float* __restrict__ V, const unsigned char* __restrict__ mask,
    float* __restrict__ out)
{
  __shared__ float tile[16 * 17];   // padded rows -> conflict-free transpose

  const int lane = threadIdx.x & 31;
  const int half = lane >> 4;       // 0: lanes 0-15, 1: lanes 16-31
  const int lq   = lane & 15;

  const int wid = blockIdx.x;       // one wave per block
  const int qb  = wid & (QB - 1);
  const int bh  = wid >> 7;         // wid / QB
  const int b   = bh / H_;
  const int h   = bh % H_;
  const bool local = (h >= GNUM);
  const int q0 = qb * 16;

  const float* __restrict__ Qb = Q + ((size_t)bh * N_ + q0) * D_;
  const float* __restrict__ Kb = K + (size_t)bh * N_ * D_;
  const float* __restrict__ Vb = V + (size_t)bh * N_ * D_;
  const unsigned char* __restrict__ mb = mask + (size_t)b * N_;

  const size_t OUT_SZ = (size_t)B_ * H_ * N_ * D_;
  const size_t PH_SZ  = (size_t)N_ * N_;
  size_t p_off = OUT_SZ;
  if (!local) p_off += (size_t)(b * GNUM + h) * PH_SZ;
  else        p_off += (size_t)B_ * GNUM * PH_SZ + (size_t)(b * GNUM + (h - GNUM)) * PH_SZ;
  float* __restrict__ P = out + p_off;                         // [N][N] for this (b,h)
  float* __restrict__ O = out + ((size_t)bh * N_ + q0) * D_;   // [16][64]

  // ---- load Q as 16 A-fragments (16x4), fold in scale 1/sqrt(D) ----
  // A layout chunk c: lane<16 holds A[lq][4c+0..1]; lane>=16 holds A[lq][4c+2..3]
  v2f qa[16];
  const float* qp = Qb + (size_t)lq * D_ + 2 * half;
  #pragma unroll
  for (int c = 0; c < 16; ++c) {
    v2f t = *(const v2f*)(qp + 4 * c);
    qa[c].x = t.x * 0.125f;
    qa[c].y = t.y * 0.125f;
  }

  int tlo = 0, thi = NT - 1;
  if (local) {                      // band |q-k| <= 64 around [q0, q0+15]
    tlo = (qb >= 4) ? qb - 4 : 0;
    thi = (qb + 4 <= NT - 1) ? qb + 4 : NT - 1;
  }

  // ================= sweep 1: row sums of exp(S) =================
  v8f rsum = {};
  for (int t = tlo; t <= thi; ++t) {
    const int k0 = t * 16;
    v8f s = {};
    // B layout chunk c: lane<16 holds B[4c+0..1][lq] = K[k0+lq][4c+0..1]; lane>=16 rows 4c+2..3
    const float* kp = Kb + (size_t)(k0 + lq) * D_ + 2 * half;
    #pragma unroll
    for (int c = 0; c < 16; ++c) {
      v2f kb = *(const v2f*)(kp + 4 * c);
      s = wmma_f32(qa[c], kb, s);
    }
    const int kcol = k0 + lq;
    const bool mk = (mb[kcol] != 0);
    #pragma unroll
    for (int r = 0; r < 8; ++r) {
      const int qrow = q0 + r + 8 * half;     // C-layout row for slot r
      int dist = qrow - kcol; if (dist < 0) dist = -dist;
      const bool ok = mk && (!local || dist <= WIN);
      rsum[r] += ok ? __expf(s[r]) : 0.0f;
    }
  }
  // reduce across the 16-lane half that shares each row (xor stays in-half)
  #pragma unroll
  for (int r = 0; r < 8; ++r) {
    float v = rsum[r];
    v += __shfl_xor(v, 1, 32);
    v += __shfl_xor(v, 2, 32);
    v += __shfl_xor(v, 4, 32);
    v += __shfl_xor(v, 8, 32);
    rsum[r] = v;
  }
  v8f rinv;
  #pragma unroll
  for (int r = 0; r < 8; ++r) rinv[r] = (rsum[r] > 0.0f) ? 1.0f / rsum[r] : 0.0f;

  // ================= sweep 2: write P once, accumulate O = P*V =================
  v8f oacc[4] = {};   // 4 column groups of 16 -> 16x64 output
  for (int t = tlo; t <= thi; ++t) {
    const int k0 = t * 16;
    v8f s = {};
    const float* kp = Kb + (size_t)(k0 + lq) * D_ + 2 * half;
    #pragma unroll
    for (int c = 0; c < 16; ++c) {
      v2f kb = *(const v2f*)(kp + 4 * c);
      s = wmma_f32(qa[c], kb, s);
    }
    const int kcol = k0 + lq;
    const bool mk = (mb[kcol] != 0);
    #pragma unroll
    for (int r = 0; r < 8; ++r) {
      const int qrow = q0 + r + 8 * half;
      int dist = qrow - kcol; if (dist < 0) dist = -dist;
      const bool ok = mk && (!local || dist <= WIN);
      const float p = ok ? __expf(s[r]) * rinv[r] : 0.0f;   // rinv[r] matches this row
      __builtin_nontemporal_store(p, &P[(size_t)qrow * N_ + kcol]); // streamed, write-once
      tile[(r + 8 * half) * 17 + lq] = p;                   // stage for C->A re-striping
    }
    __syncthreads();   // single-wave WG: S_NOP + compiler fence; compiler emits s_wait_dscnt

    // read P tile back as A-fragments (K-dim = key index)
    v2f pa[4];
    #pragma unroll
    for (int c = 0; c < 4; ++c) {
      pa[c].x = tile[lq * 17 + 4 * c + 2 * half];
      pa[c].y = tile[lq * 17 + 4 * c + 2 * half + 1];
    }
    // O[16x64] += P[16x16] * V[16x64], 4 chunks (K) x 4 groups (N)
    #pragma unroll
    for (int g = 0; g < 4; ++g) {
      #pragma unroll
      for (int c = 0; c < 4; ++c) {
        const int kr = k0 + 4 * c + 2 * half;
        v2f vb;
        vb.x = Vb[(size_t)kr * D_ + 16 * g + lq];
        vb.y = Vb[(size_t)(kr + 1) * D_ + 16 * g + lq];
        oacc[g] = wmma_f32(pa[c], vb, oacc[g]);
      }
    }
    __syncthreads();
  }

  // store output tile (rows of oacc C-fragments == rows of rinv, already normalized)
  #pragma unroll
  for (int g = 0; g < 4; ++g) {
    #pragma unroll
    for (int r = 0; r < 8; ++r) {
      O[(size_t)(r + 8 * half) * D_ + 16 * g + lq] = oacc[g][r];
    }
  }
}

extern "C" void kernel_launch(void* const* d_in, const int* in_sizes, int n_in,
                              void* d_out, int out_size, void* d_ws, size_t ws_size,
                              hipStream_t stream) {
  const float*         Q    = (const float*)d_in[0];
  const float*         K    = (const float*)d_in[1];
  const float*         V    = (const float*)d_in[2];
  const unsigned char* mask = (const unsigned char*)d_in[3];  // jax bool: 1 byte/elem
  float* out = (float*)d_out;

  const size_t OUT_SZ = (size_t)B_ * H_ * N_ * D_;       // 4,194,304
  const size_t PG_SZ  = (size_t)B_ * GNUM * N_ * N_;     // 67,108,864

  // p_l is mostly out-of-band zeros: bulk-zero it (capturable), kernel fills the band.
  hipMemsetAsync(out + OUT_SZ + PG_SZ, 0, PG_SZ * sizeof(float), stream);

  // one wave32 per (b, h, 16-query block)
  const int nblocks = B_ * H_ * QB;   // 4096
  wattn_kernel<<<nblocks, 32, 0, stream>>>(Q, K, V, mask, out);
}